// BertForMultiHopQuestionAnswering_23630910063160
// MI455X (gfx1250) — compile-verified
//
#include <hip/hip_runtime.h>
#include <hip/hip_bf16.h>
#include <cstdint>

typedef __attribute__((ext_vector_type(2))) float v2f;
typedef __attribute__((ext_vector_type(8))) float v8f;

#define BB   64
#define LL   512
#define HH   1024
#define SS   20
#define ROWS (BB * LL)          // 32768 flattened (b,l) rows
#define MAX_SPAN_C 10

// ---------------------------------------------------------------------------
// Kernel 1: logits[b,l,0:4] = seq[b,l,:] @ qa_w + qa_b  via V_WMMA_F32_16X16X4_F32
// One wave owns a 32-row tile (two 16x16x4 WMMA chains sharing one B fragment).
// B staged in LDS transposed + zero-padded: row n' (0..3) = qa_w column n',
// row 4 = zeros for lanes with n>=4. Inner loop is branch/EXEC-mask free:
//   2x global_load_b64 (A) + 1x ds_load_b64 (B) + 2x v_wmma per k-step.
// ---------------------------------------------------------------------------
__global__ void __launch_bounds__(128)
qa_logits_wmma_kernel(const float* __restrict__ seq,   // [ROWS, HH]
                      const float* __restrict__ qa_w,  // [HH, 4]
                      const float* __restrict__ qa_b,  // [4]
                      float* __restrict__ logits)      // [ROWS, 4]
{
    __shared__ float lds_b[5 * HH];                    // 20 KB of 320 KB/WGP
    const int tid = threadIdx.x;
    for (int i = tid; i < 5 * HH; i += blockDim.x) {
        const int n = i >> 10;                         // 0..4
        const int h = i & (HH - 1);
        lds_b[i] = (n < 4) ? qa_w[h * 4 + n] : 0.0f;   // row 4 = zero pad
    }
    __syncthreads();

    const int wave = tid >> 5;
    const int lane = tid & 31;
    const int tile = blockIdx.x * 4 + wave;            // ROWS/32 = 1024 tiles
    const int R    = tile * 32;

    const int m  = lane & 15;          // A row within tile / output column idx
    const int kb = (lane >> 4) << 1;   // half-wave K offset: 0 or 2
    const int n  = lane & 15;          // B/output column (only n<4 real)
    const int n5 = (n < 4) ? n : 4;    // n>=4 lanes -> zero row (broadcast)

    const float* arow0 = seq + (size_t)(R + m) * HH + kb;
    const float* arow1 = arow0 + (size_t)16 * HH;
    const float* brow  = lds_b + n5 * HH + kb;

    v8f c0 = (v8f)0.0f;
    v8f c1 = (v8f)0.0f;
    #pragma unroll 4
    for (int k = 0; k < HH; k += 4) {
        // A 16x4 fp32: lane holds A[m][k+kb], A[m][k+kb+1] (contiguous 8B)
        v2f a0 = *(const v2f*)(arow0 + k);
        v2f a1 = *(const v2f*)(arow1 + k);
        // B 4x16 fp32: lane holds B[k+kb][n], B[k+kb+1][n] (contiguous 8B in LDS)
        v2f bv = *(const v2f*)(brow + k);
        c0 = __builtin_amdgcn_wmma_f32_16x16x4_f32(
                false, a0, false, bv, (short)0, c0, false, false);
        c1 = __builtin_amdgcn_wmma_f32_16x16x4_f32(
                false, a1, false, bv, (short)0, c1, false, false);
    }

    // C/D 16x16 f32 layout: lane l, vgpr r -> (M = r + 8*(l>>4), N = l&15)
    if (n < 4) {
        const float bias  = qa_b[n];
        const int   mbase = (lane >> 4) * 8;
        float* o0 = logits + (size_t)(R + mbase) * 4 + n;
        float* o1 = o0 + 16 * 4;
        #pragma unroll
        for (int r = 0; r < 8; ++r) {
            o0[r * 4] = c0[r] + bias;
            o1[r * 4] = c1[r] + bias;
        }
    }
}

// ---------------------------------------------------------------------------
// Kernel 2: semantics = seq[:, 0, :]   (B x H strided copy)
// ---------------------------------------------------------------------------
__global__ void semantics_copy_kernel(const float* __restrict__ seq,
                                      float* __restrict__ out)  // [BB*HH]
{
    const int i = blockIdx.x * blockDim.x + threadIdx.x;
    if (i >= BB * HH) return;
    const int b = i >> 10;
    const int h = i & (HH - 1);
    out[i] = seq[(size_t)b * LL * HH + h];
}

// ---------------------------------------------------------------------------
// Kernel 3: span extraction. One wave32 per (sample b, mode).
//   mode 0: hop (K=3, logit cols 0/1) -> hop_preds
//   mode 1: ans (K=1, logit cols 2/3) -> ans_preds + ans_start_gap
// ---------------------------------------------------------------------------
__global__ void __launch_bounds__(32)
extract_kernel(const float* __restrict__ logits,   // [BB, LL, 4]
               const int* __restrict__ seps,       // [BB, SS]
               const int* __restrict__ bstarts,    // [BB]
               float* __restrict__ out)            // full output buffer
{
    const int b    = blockIdx.x;
    const int mode = blockIdx.y;           // 0=hop, 1=ans
    const int K    = mode ? 1 : 3;
    const int cs   = mode ? 2 : 0;
    const int ce   = cs + 1;
    const int lane = threadIdx.x;

    const float* lg = logits + (size_t)b * LL * 4;
    const int*   sp = seps + b * SS;
    const int bstart = bstarts[b];

    const float sl0    = lg[cs];           // start_logits[0]
    const float thresh = sl0;              // allow == 0

    float vals[3];
    int   starts[3];
    int ex0 = -1, ex1 = -1;                // previously-selected indices

    for (int kk = 0; kk < K; ++kk) {
        float best = -__builtin_inff();
        int   bidx = 0x7fffffff;
        for (int i = lane; i < LL; i += 32) {
            if (i == ex0 || i == ex1) continue;
            float v = (i >= bstart) ? lg[i * 4 + cs] : -__builtin_inff();
            if (v > best || (v == best && i < bidx)) { best = v; bidx = i; }
        }
        #pragma unroll
        for (int off = 16; off > 0; off >>= 1) {
            float ov = __shfl_xor(best, off, 32);
            int   oi = __shfl_xor(bidx, off, 32);
            if (ov > best || (ov == best && oi < bidx)) { best = ov; bidx = oi; }
        }
        vals[kk]   = best;
        starts[kk] = bidx;
        if (kk == 0) ex0 = bidx; else ex1 = bidx;
    }

    if (lane != 0) return;

    int okf[3], endv[3], jv[3];
    for (int kk = 0; kk < K; ++kk) {
        const int st = starts[kk];
        // j = first s with (seps[s] > st) || (seps[s] <= 0), else S-1
        int j = SS - 1;
        for (int s = 0; s < SS; ++s)
            if (sp[s] > st || sp[s] <= 0) { j = s; break; }
        const int ending = sp[j];
        const int cond1  = vals[kk] > thresh;
        const int cond2  = ending > st;
        okf[kk] = cond1 && cond2;
        // end = st + argmax of end_logits over window [st, min(ending, st+10))
        const int cap = (ending < st + MAX_SPAN_C) ? ending : st + MAX_SPAN_C;
        float bw = -__builtin_inff();
        int   bt = 0;
        for (int t = 0; t < MAX_SPAN_C; ++t) {
            const int pos = st + t;
            int cp = pos;
            if (cp < 0) cp = 0;
            if (cp > LL - 1) cp = LL - 1;
            const float w = (pos < cap) ? lg[cp * 4 + ce] : -__builtin_inff();
            if (w > bw) { bw = w; bt = t; }   // strict > : tie -> lowest t
        }
        endv[kk] = st + bt;
        jv[kk]   = j;
    }

    const int active = sp[0] > 0;
    // output layout (floats): hop_preds[64,3,3] | ans_preds[64,1,3] |
    //                         semantics[64,1024] | ans_start_gap[64]
    float* pred_out = mode ? (out + BB * 9 + b * 3) : (out + b * 9);
    int run = 1;
    for (int kk = 0; kk < K; ++kk) {
        run = run && okf[kk];
        const int v = run && active;
        pred_out[kk * 3 + 0] = v ? (float)starts[kk] : 0.0f;
        pred_out[kk * 3 + 1] = v ? (float)endv[kk]   : 0.0f;
        pred_out[kk * 3 + 2] = v ? (float)jv[kk]     : 0.0f;
    }

    if (mode == 1) {
        int any_bad = 0, fb = 0;
        for (int kk = 0; kk < K; ++kk)
            if (!okf[kk]) { any_bad = 1; fb = kk; break; }
        float gap = 0.0f;
        if (active && any_bad && !(vals[fb] > thresh))
            gap = sl0 - vals[fb];
        out[BB * 9 + BB * 3 + BB * HH + b] = gap;
    }
}

// ---------------------------------------------------------------------------
extern "C" void kernel_launch(void* const* d_in, const int* in_sizes, int n_in,
                              void* d_out, int out_size, void* d_ws, size_t ws_size,
                              hipStream_t stream) {
    const float* seq   = (const float*)d_in[0];   // [64,512,1024] f32
    const float* qa_w  = (const float*)d_in[1];   // [1024,4] f32
    const float* qa_b  = (const float*)d_in[2];   // [4] f32
    const int*   seps  = (const int*)d_in[3];     // [64,20] i32
    const int*   bst   = (const int*)d_in[4];     // [64] i32
    // d_in[5..8] (hop/ans loss weights) unused by the reference.

    float* logits = (float*)d_ws;                 // 32768*4 f32 = 512 KB scratch
    float* outf   = (float*)d_out;

    // 1024 tiles of 32 rows; 4 waves (tiles) per 128-thread block
    qa_logits_wmma_kernel<<<ROWS / 32 / 4, 128, 0, stream>>>(seq, qa_w, qa_b, logits);

    semantics_copy_kernel<<<(BB * HH + 255) / 256, 256, 0, stream>>>(
        seq, outf + BB * 9 + BB * 3);

    extract_kernel<<<dim3(BB, 2, 1), 32, 0, stream>>>(logits, seps, bst, outf);
}